// SAGE_24842090840540
// MI455X (gfx1250) — compile-verified
//
#include <hip/hip_runtime.h>
#include <math.h>

// GraphSAGE 2-layer, fp32. Bottleneck = L2-resident edge scatter (x fits in
// 192MB L2). GEMMs via V_WMMA_F32_16X16X4_F32 (full fp32 precision).
// Weights staged in LDS with k-pair interleaving so each B fragment is a
// single aligned ds_load_b64 (no VGPR re-pairing movs in the WMMA loop).

#define NF 128
#define BN_EPS 1e-5f

typedef __attribute__((ext_vector_type(2))) float v2f;
typedef __attribute__((ext_vector_type(8))) float v8f;

__global__ void k_zero(float* __restrict__ p, long n) {
  long i = (long)blockIdx.x * blockDim.x + threadIdx.x;
  if (i < n) p[i] = 0.0f;
}

__global__ void k_degree(const int* __restrict__ dst, float* __restrict__ deg, int nE) {
  int e = blockIdx.x * blockDim.x + threadIdx.x;
  if (e < nE) atomicAdd(&deg[dst[e]], 1.0f);
}

__global__ void k_invdeg(float* __restrict__ deg, int n) {
  int i = blockIdx.x * blockDim.x + threadIdx.x;
  if (i < n) deg[i] = 1.0f / fmaxf(deg[i], 1.0f);
}

// One wave32 per edge: lane l moves features [4l, 4l+3] (float4 = 512B/row/wave).
__global__ void k_scatter(const float* __restrict__ feat,
                          const int* __restrict__ src,
                          const int* __restrict__ dst,
                          float* __restrict__ agg, int nE) {
  int t = blockIdx.x * blockDim.x + threadIdx.x;
  int e = t >> 5;
  int lane = t & 31;
  if (e >= nE) return;
  int s = src[e];
  int d = dst[e];
  const float4 v = ((const float4*)(feat + (size_t)s * NF))[lane];
  float* o = agg + (size_t)d * NF + lane * 4;
  atomicAdd(o + 0, v.x);
  atomicAdd(o + 1, v.y);
  atomicAdd(o + 2, v.z);
  atomicAdd(o + 3, v.w);
}

// Fused SAGE layer: out[16x128 block] = (agg*invdeg) @ Wl^T + bias + x @ Wr^T,
// then per-row L2 normalize (+ optional ReLU). One wave per 16-row block.
// WMMA f32 16x16x4 layouts (ISA 7.12.2):
//   A: lane<16 holds A[lid][k0],A[lid][k0+1]; lane>=16 holds A[lid][k0+2],A[lid][k0+3]
//   B: lane<16 holds B[k0][lid],B[k0+1][lid]; lane>=16 holds B[k0+2][lid],B[k0+3][lid]
//   C/D: VGPR v, lanes0-15 -> (row v, col lid), lanes16-31 -> (row v+8, col lid)
// LDS weight layout (k-pair interleaved): W2[k>>1][j][k&1] at (k>>1)*256 + j*2 + (k&1),
// so the lane's B fragment (WT[kk][j], WT[kk+1][j]) is one aligned float2.
__global__ void __launch_bounds__(256) k_sage_wmma(
    const float* __restrict__ agg, const float* __restrict__ invdeg,
    const float* __restrict__ xroot,
    const float* __restrict__ Wl, const float* __restrict__ bl,
    const float* __restrict__ Wr,
    float* __restrict__ out, int do_relu, int nRows) {
  extern __shared__ float lds[];  // [0,16384): Wl^T pairs; [16384,32768): Wr^T pairs

  for (int i = threadIdx.x; i < NF * NF; i += blockDim.x) {
    int j = i >> 7;          // output feature (row of W)
    int k = i & (NF - 1);    // input feature  (col of W)
    int a = (k >> 1) * (2 * NF) + j * 2 + (k & 1);
    lds[a] = Wl[i];
    lds[NF * NF + a] = Wr[i];
  }
  __syncthreads();

  const int lane = threadIdx.x & 31;
  const int halfsel = lane >> 4;   // 0: lanes 0-15, 1: lanes 16-31
  const int lid = lane & 15;
  const int wavesPerBlock = blockDim.x >> 5;
  const int gwave = blockIdx.x * wavesPerBlock + (threadIdx.x >> 5);
  const int nWaves = gridDim.x * wavesPerBlock;
  const int nRB = nRows >> 4;      // N = 100000 is divisible by 16

  // Per-lane LDS base for B fragments: kk2 = kt*2 + halfsel, col = jt*16 + lid
  // addr(kt, jt) = (kt*2 + halfsel)*256 + (jt*16 + lid)*2   [floats]
  const v2f* __restrict__ Bl2 = (const v2f*)(lds) + (halfsel * 128 + lid);
  const v2f* __restrict__ Br2 = (const v2f*)(lds + NF * NF) + (halfsel * 128 + lid);

  // Bias is row-block invariant: preload the 8 per-column values.
  float bias[8];
#pragma unroll
  for (int jt = 0; jt < 8; ++jt) bias[jt] = bl[jt * 16 + lid];

  for (int rb = gwave; rb < nRB; rb += nWaves) {
    const int m0 = rb << 4;
    const int arow = m0 + lid;     // A-matrix row owned by this lane
    const float idg = invdeg[arow];

    v8f acc[8];
#pragma unroll
    for (int jt = 0; jt < 8; ++jt)
#pragma unroll
      for (int v = 0; v < 8; ++v) acc[jt][v] = bias[jt];

    // acc += (agg * invdeg) @ Wl^T
    const float2* A2 = (const float2*)(agg + (size_t)arow * NF) + halfsel;
    for (int kt = 0; kt < 32; ++kt) {
      const float2 af = A2[kt * 2];
      v2f a;
      a.x = af.x * idg;
      a.y = af.y * idg;
#pragma unroll
      for (int jt = 0; jt < 8; ++jt) {
        const v2f b = Bl2[kt * 256 + jt * 16];  // one ds_load_b64
        acc[jt] = __builtin_amdgcn_wmma_f32_16x16x4_f32(
            false, a, false, b, (short)0, acc[jt], false, false);
      }
    }

    // acc += x @ Wr^T
    const float2* X2 = (const float2*)(xroot + (size_t)arow * NF) + halfsel;
    for (int kt = 0; kt < 32; ++kt) {
      const float2 af = X2[kt * 2];
      v2f a;
      a.x = af.x;
      a.y = af.y;
#pragma unroll
      for (int jt = 0; jt < 8; ++jt) {
        const v2f b = Br2[kt * 256 + jt * 16];
        acc[jt] = __builtin_amdgcn_wmma_f32_16x16x4_f32(
            false, a, false, b, (short)0, acc[jt], false, false);
      }
    }

    // Row L2 norm: row of acc[jt][v] is (m0 + v + 8*halfsel); its 128 columns
    // live across the 16 lanes of this half-wave over the 8 jt tiles.
    float nrm[8];
#pragma unroll
    for (int v = 0; v < 8; ++v) nrm[v] = 0.0f;
#pragma unroll
    for (int jt = 0; jt < 8; ++jt)
#pragma unroll
      for (int v = 0; v < 8; ++v) nrm[v] += acc[jt][v] * acc[jt][v];
#pragma unroll
    for (int v = 0; v < 8; ++v) {
#pragma unroll
      for (int m = 1; m < 16; m <<= 1)  // xor 1,2,4,8 stays within half-wave
        nrm[v] += __shfl_xor(nrm[v], m, 32);
      nrm[v] = 1.0f / fmaxf(sqrtf(nrm[v]), 1e-12f);
    }

#pragma unroll
    for (int jt = 0; jt < 8; ++jt) {
#pragma unroll
      for (int v = 0; v < 8; ++v) {
        float val = acc[jt][v] * nrm[v];
        if (do_relu) val = fmaxf(val, 0.0f);
        const int row = m0 + v + 8 * halfsel;
        out[(size_t)row * NF + jt * 16 + lid] = val;
      }
    }
  }
}

// Per-feature sum / sum-of-squares partials (blockDim.x == 128).
__global__ void k_bn_stats(const float* __restrict__ h, float* __restrict__ sums, int n) {
  const int f = threadIdx.x;
  const int rowsPer = (n + gridDim.x - 1) / gridDim.x;
  const int r0 = blockIdx.x * rowsPer;
  const int r1 = min(n, r0 + rowsPer);
  float s = 0.0f, s2 = 0.0f;
  for (int r = r0; r < r1; ++r) {
    float v = h[(size_t)r * NF + f];
    s += v;
    s2 += v * v;
  }
  atomicAdd(&sums[f], s);
  atomicAdd(&sums[NF + f], s2);
}

__global__ void k_bn_apply(float* __restrict__ h, const float* __restrict__ sums,
                           const float* __restrict__ gamma, const float* __restrict__ beta,
                           long n, float invN) {
  long i = (long)blockIdx.x * blockDim.x + threadIdx.x;
  if (i >= n) return;
  int f = (int)(i & (NF - 1));
  float mu = sums[f] * invN;
  float var = sums[NF + f] * invN - mu * mu;
  h[i] = (h[i] - mu) * rsqrtf(var + BN_EPS) * gamma[f] + beta[f];
}

extern "C" void kernel_launch(void* const* d_in, const int* in_sizes, int n_in,
                              void* d_out, int out_size, void* d_ws, size_t ws_size,
                              hipStream_t stream) {
  const float* x     = (const float*)d_in[0];
  const int*   ei    = (const int*)d_in[1];   // [2,E] int32 (JAX default x64-off)
  const float* Wl1   = (const float*)d_in[2];
  const float* bl1   = (const float*)d_in[3];
  const float* Wr1   = (const float*)d_in[4];
  const float* gamma = (const float*)d_in[5];
  const float* beta  = (const float*)d_in[6];
  const float* Wl2   = (const float*)d_in[7];
  const float* bl2   = (const float*)d_in[8];
  const float* Wr2   = (const float*)d_in[9];
  float* out = (float*)d_out;

  const int nE = in_sizes[1] / 2;
  const int n  = in_sizes[0] / NF;
  const size_t nf = (size_t)n * NF;

  float* ws   = (float*)d_ws;
  float* agg  = ws;                 // [n,128]
  float* h    = ws + nf;            // [n,128]
  float* deg  = ws + 2 * nf;        // [n] -> becomes invdeg
  float* sums = ws + 2 * nf + n;    // [256]

  const int T = 256;
  const int* src = ei;
  const int* dst = ei + nE;
  const long sthreads = (long)nE * 32;
  const int sblocks = (int)((sthreads + T - 1) / T);
  const size_t ldsBytes = 2 * NF * NF * sizeof(float);  // 128 KB

  // ---- Layer 1 ----
  k_zero<<<(int)((nf + T - 1) / T), T, 0, stream>>>(agg, (long)nf);
  k_zero<<<(n + 256 + T - 1) / T, T, 0, stream>>>(deg, (long)(n + 256));
  k_degree<<<(nE + T - 1) / T, T, 0, stream>>>(dst, deg, nE);
  k_invdeg<<<(n + T - 1) / T, T, 0, stream>>>(deg, n);
  k_scatter<<<sblocks, T, 0, stream>>>(x, src, dst, agg, nE);
  k_sage_wmma<<<512, 256, ldsBytes, stream>>>(agg, deg, x, Wl1, bl1, Wr1, h,
                                              /*do_relu=*/1, n);
  // ---- ReLU already fused; BatchNorm (batch stats) ----
  k_bn_stats<<<512, 128, 0, stream>>>(h, sums, n);
  k_bn_apply<<<(int)((nf + T - 1) / T), T, 0, stream>>>(h, sums, gamma, beta,
                                                        (long)nf, 1.0f / (float)n);
  // ---- Layer 2 ----
  k_zero<<<(int)((nf + T - 1) / T), T, 0, stream>>>(agg, (long)nf);
  k_scatter<<<sblocks, T, 0, stream>>>(h, src, dst, agg, nE);
  k_sage_wmma<<<512, 256, ldsBytes, stream>>>(agg, deg, h, Wl2, bl2, Wr2, out,
                                              /*do_relu=*/0, n);
}